// SparseBMM_89421219103268
// MI455X (gfx1250) — compile-verified
//
#include <hip/hip_runtime.h>

typedef __attribute__((ext_vector_type(2))) float v2f;
typedef __attribute__((ext_vector_type(4))) float v4f;
typedef __attribute__((ext_vector_type(8))) float v8f;

#define BATCH 16
#define M_DIM 2048
#define K_DIM 2048
#define N_DIM 64
#define KTILE 32
#define MTILE 32
#define LDA   33   // padded LDS row stride for A (33 mod 32 banks = conflict-free columns)

__global__ __launch_bounds__(128) void sparse_bmm_wmma(const float* __restrict__ A,
                                                       const float* __restrict__ Bm,
                                                       float* __restrict__ C) {
  __shared__ float sA[MTILE * LDA];     // 32x32 A chunk, padded
  __shared__ float sB[KTILE * N_DIM];   // 32x64 B chunk
  __shared__ int   s_nz;

  const int tid   = threadIdx.x;
  const int lane  = tid & 31;
  const int wave  = tid >> 5;            // 0..3 -> 16-col slice of N
  const int m16   = lane & 15;
  const int khalf = (lane < 16) ? 0 : 2; // A/B lane-half K offset per ISA layout
  const int batch = blockIdx.y;
  const int m0    = blockIdx.x * MTILE;
  const int n0    = wave * 16;

  const float* Ab = A  + ((size_t)batch * M_DIM + m0) * K_DIM;
  const float* Bb = Bm + (size_t)batch * K_DIM * N_DIM;

  // A staging: thread t -> row t/4, 8 floats at col (t%4)*8 (128B/contig per 4 threads)
  const int arow = tid >> 2;
  const int acol = (tid & 3) * 8;
  // B staging: thread t -> row t/4, 16 floats at col (t%4)*16
  const int brow = tid >> 2;
  const int bcol = (tid & 3) * 16;

  v8f acc0 = {};  // C rows m0..m0+15,  cols n0..n0+15
  v8f acc1 = {};  // C rows m0+16..+31, cols n0..n0+15

  for (int kc = 0; kc < K_DIM; kc += KTILE) {
    if (tid == 0) s_nz = 0;

    // ---- stream the 32x32 A chunk (two global_load_b128 per thread) ----
    const float* ap = Ab + (size_t)arow * K_DIM + kc + acol;
    v4f a0 = *(const v4f*)(ap);
    v4f a1 = *(const v4f*)(ap + 4);

    if (kc + KTILE < K_DIM) {
      __builtin_prefetch(ap + KTILE, 0, 0);   // global_prefetch_b8: keep A stream deep
    }

    // zero-tile detection on raw bits (A chunk == one sparsity tile)
    unsigned nzb = 0;
#pragma unroll
    for (int i = 0; i < 4; ++i) {
      nzb |= __float_as_uint(a0[i]) | __float_as_uint(a1[i]);
    }

    // stage A to padded LDS rows
    float* as = &sA[arow * LDA + acol];
#pragma unroll
    for (int i = 0; i < 4; ++i) { as[i] = a0[i]; as[4 + i] = a1[i]; }

    __syncthreads();            // sA stores + s_nz reset visible
    if (nzb) s_nz = 1;          // benign race: all writers store 1
    __syncthreads();            // flag settled
    const int nz = s_nz;        // uniform across workgroup -> EXEC stays full

    if (nz) {
      // ---- stage the 32x64 B chunk ----
      const float* bp = Bb + (size_t)(kc + brow) * N_DIM + bcol;
      v4f b0 = *(const v4f*)(bp);
      v4f b1 = *(const v4f*)(bp + 4);
      v4f b2 = *(const v4f*)(bp + 8);
      v4f b3 = *(const v4f*)(bp + 12);
      v4f* bs = (v4f*)&sB[brow * N_DIM + bcol];
      bs[0] = b0; bs[1] = b1; bs[2] = b2; bs[3] = b3;
      __syncthreads();

      // ---- 8 x (16x16x4 fp32 WMMA) per accumulator ----
#pragma unroll
      for (int kk = 0; kk < KTILE; kk += 4) {
        const float* ac0 = &sA[m16 * LDA + kk + khalf];
        const float* ac1 = &sA[(m16 + 16) * LDA + kk + khalf];
        v2f fa0; fa0.x = ac0[0]; fa0.y = ac0[1];
        v2f fa1; fa1.x = ac1[0]; fa1.y = ac1[1];
        const float* bq = &sB[(kk + khalf) * N_DIM + n0 + m16];
        v2f fb;  fb.x = bq[0];  fb.y = bq[N_DIM];

        acc0 = __builtin_amdgcn_wmma_f32_16x16x4_f32(false, fa0, false, fb,
                                                     (short)0, acc0, false, false);
        acc1 = __builtin_amdgcn_wmma_f32_16x16x4_f32(false, fa1, false, fb,
                                                     (short)0, acc1, false, false);
      }
    }
    __syncthreads();            // protect sA/sB/s_nz before next chunk
  }

  // ---- write 32x64 C tile (C/D layout: VGPR r -> M = r + (lane<16?0:8)) ----
  const int mrow = (lane < 16) ? 0 : 8;
  float* cp = C + ((size_t)batch * M_DIM + m0) * N_DIM + n0 + m16;
#pragma unroll
  for (int r = 0; r < 8; ++r) {
    cp[(size_t)(mrow + r) * N_DIM]      = acc0[r];
    cp[(size_t)(16 + mrow + r) * N_DIM] = acc1[r];
  }
}

extern "C" void kernel_launch(void* const* d_in, const int* in_sizes, int n_in,
                              void* d_out, int out_size, void* d_ws, size_t ws_size,
                              hipStream_t stream) {
  (void)in_sizes; (void)n_in; (void)out_size; (void)d_ws; (void)ws_size;
  const float* A  = (const float*)d_in[0];
  const float* Bm = (const float*)d_in[1];
  float*       C  = (float*)d_out;
  dim3 grid(M_DIM / MTILE, BATCH);   // 64 x 16 = 1024 workgroups
  sparse_bmm_wmma<<<grid, 128, 0, stream>>>(A, Bm, C);
}